// WaveletModel_15101105013142
// MI455X (gfx1250) — compile-verified
//
#include <hip/hip_runtime.h>

// ---------------------------------------------------------------------------
// WaveletModel on gfx1250: depthwise "conv1d same" with kernel length N cast
// as batched 16x16 fp32 WMMA GEMMs (V_WMMA_F32_16X16X4_F32).
//   out[b,k,t] = sum_j x[b, k%8, t-1151+j] * wav[k%16, j]
// Only 16 distinct rows per batch; each replicated to 8 output channels.
// ---------------------------------------------------------------------------

typedef __attribute__((ext_vector_type(2))) float v2f;
typedef __attribute__((ext_vector_type(4))) float v4f;
typedef __attribute__((ext_vector_type(8))) float v8f;

#define B_        8
#define C_        8
#define S_        16
#define N_        2304
#define PAD_L     1151          // (N-1)/2
#define XPAD_LEN  4608          // PAD_L + N + PAD_R (1152) rounded: 1151+2304+1152=4607 -> 4608
#define T_TILES   144           // N / 16
#define K4        576           // N / 4  (K-steps of 4 for f32 WMMA)

__device__ __forceinline__ float wavelet_val(float scale, int j) {
    const float two_pi = 6.28318530717958647692f;
    float sigma = scale / two_pi;
    float s2 = sigma * sigma;
    float t2 = (float)j * (float)j;
    float r  = t2 / s2;
    return (1.0f - r) * expf(-0.5f * r);
}

// wavB[jb4*64 + 2*lane + q] = wav[lane%16, 4*jb4 + (lane<16?0:2) + q]
// i.e. exactly the per-lane B fragment for WMMA iteration jb4, so the main
// loop's B load is a single coalesced, wave-uniform global_load_b64.
__global__ void build_wavB(const float* __restrict__ scales, float* __restrict__ wavB) {
    int e = blockIdx.x * blockDim.x + threadIdx.x;
    if (e >= K4 * 64) return;
    int jb4 = e >> 6;
    int r   = e & 63;
    int l   = r >> 1;     // lane 0..31
    int q   = r & 1;
    int s   = l & 15;
    int j   = 4 * jb4 + ((l < 16) ? 0 : 2) + q;
    wavB[e] = wavelet_val(scales[s], j);
}

// Zero-padded x: xpad[bc][i] = x[bc][i - PAD_L] for i in [PAD_L, PAD_L+N), else 0.
__global__ void build_xpad(const float* __restrict__ x, float* __restrict__ xpad) {
    int e = blockIdx.x * blockDim.x + threadIdx.x;
    if (e >= B_ * C_ * XPAD_LEN) return;
    int i  = e % XPAD_LEN;
    int bc = e / XPAD_LEN;
    int u  = i - PAD_L;
    xpad[e] = (u >= 0 && u < N_) ? x[bc * N_ + u] : 0.0f;
}

// One wave per (b, c, t-tile): D[16t x 16s] += A(Toeplitz x) * B(wavelets).
__global__ __launch_bounds__(256) void wavelet_wmma(
        const float* __restrict__ xpad,
        const float* __restrict__ wavB,
        float* __restrict__ out) {
    __shared__ float tile[8][16][16];   // [wave][t_local][scale]

    const int lane = threadIdx.x & 31;
    const int wlid = threadIdx.x >> 5;
    const int tid  = blockIdx.x * 8 + wlid;         // 0..9215
    const int t_tile = tid % T_TILES;
    const int c      = (tid / T_TILES) & (C_ - 1);
    const int b      = tid / (T_TILES * C_);
    const int t0     = t_tile * 16;

    // A fragment source: lane L holds A[M=L%16, K = klo + {0,1}], klo = (L<16)?0:2
    const float* __restrict__ Arow = xpad
        + (size_t)(b * C_ + c) * XPAD_LEN + t0
        + (lane & 15) + ((lane < 16) ? 0 : 2);
    // B fragment source: pre-swizzled, 8B per lane per K-step
    const float* __restrict__ Brow = wavB + lane * 2;

    v8f acc = {};
#pragma unroll 4
    for (int jb4 = 0; jb4 < K4; ++jb4) {
        v2f a;
        a.x = Arow[jb4 * 4];
        a.y = Arow[jb4 * 4 + 1];
        v2f bb = *(const v2f*)(Brow + jb4 * 64);
        // D = A(16x4 f32) * B(4x16 f32) + C
        acc = __builtin_amdgcn_wmma_f32_16x16x4_f32(
            false, a, false, bb, (short)0, acc, false, false);
    }

    // Stage D to LDS: lane holds column s = lane%16, rows M = v + 8*(lane>=16)
    const int scol  = lane & 15;
    const int mbase = (lane >> 4) * 8;
#pragma unroll
    for (int v = 0; v < 8; ++v)
        tile[wlid][mbase + v][scol] = acc[v];
    __syncthreads();

    // Valid columns of this tile: s = c and s = c+8. Each feeds 8 output
    // channels k = s + 16*m (k%16 == s, k%8 == c). 256 floats per wave:
    // lane -> (channel index lane>>1 of 16, t-half lane&1), 32B b128 stores.
    const int ci    = lane >> 1;
    const int s_sel = ci >> 3;
    const int m     = ci & 7;
    const int s     = c + 8 * s_sel;
    const int k     = s + 16 * m;
    const int th    = (lane & 1) * 8;
    float* __restrict__ dst = out + ((size_t)(b * (C_ * S_) + k)) * N_ + t0 + th;
    v4f o0, o1;
#pragma unroll
    for (int q = 0; q < 4; ++q) o0[q] = tile[wlid][th + q][s];
#pragma unroll
    for (int q = 0; q < 4; ++q) o1[q] = tile[wlid][th + 4 + q][s];
    *(v4f*)(dst)     = o0;
    *(v4f*)(dst + 4) = o1;
}

extern "C" void kernel_launch(void* const* d_in, const int* in_sizes, int n_in,
                              void* d_out, int out_size, void* d_ws, size_t ws_size,
                              hipStream_t stream) {
    const float* x      = (const float*)d_in[0];   // [8,8,48,48] fp32
    const float* scales = (const float*)d_in[1];   // [16] fp32
    (void)d_in; (void)in_sizes; (void)n_in; (void)ws_size;

    float* wavB = (float*)d_ws;                                    // 576*64 floats = 147456 B
    float* xpad = (float*)((char*)d_ws + (size_t)K4 * 64 * 4);     // 64*4608 floats = 1.18 MB

    {   // wavelet table, lane-swizzled for WMMA B fragments
        int n = K4 * 64;
        build_wavB<<<(n + 255) / 256, 256, 0, stream>>>(scales, wavB);
    }
    {   // zero-padded input
        int n = B_ * C_ * XPAD_LEN;
        build_xpad<<<(n + 255) / 256, 256, 0, stream>>>(x, xpad);
    }
    {   // 9216 tiles, 8 waves (one tile each) per 256-thread block
        wavelet_wmma<<<(B_ * C_ * T_TILES) / 8, 256, 0, stream>>>(
            xpad, wavB, (float*)d_out);
    }
}